// ScreenedCoulombEnergy_49563922596532
// MI455X (gfx1250) — compile-verified
//
#include <hip/hip_runtime.h>

#define RADIUS_F 5.0f
#define ECONV_F  14.399645f
#define PI_F     3.14159265358979323846f

typedef float v4f __attribute__((ext_vector_type(4)));
typedef int   v4i __attribute__((ext_vector_type(4)));

// Address-space-qualified vector types for the async global->LDS DMA builtin:
// param0 is v4i in addrspace(1) (global), param1 is v4i in addrspace(3) (LDS).
typedef __attribute__((address_space(1))) v4i as1_v4i;
typedef __attribute__((address_space(3))) v4i as3_v4i;

__global__ __launch_bounds__(1024, 1)
void screened_coulomb_kernel(const float* __restrict__ charges,
                             const float* __restrict__ pair_dist,
                             const int*   __restrict__ pair_first,
                             const int*   __restrict__ pair_second,
                             const int*   __restrict__ mol_index,
                             float* __restrict__ out,
                             int n_pairs, int n_atoms, int n_mol)
{
  extern __shared__ float lds[];
  float* lds_q   = lds;            // n_atoms floats (256 KB): whole charge table
  float* lds_bin = lds + n_atoms;  // n_mol floats (8 KB): molecule bins

  const int tid  = (int)threadIdx.x;
  const int nthr = (int)blockDim.x;

  // zero the per-workgroup molecule bins
  for (int b = tid; b < n_mol; b += nthr) lds_bin[b] = 0.0f;

  // ---- Stage the charge table into LDS (async DMA global -> LDS) ----
  const v4f* q4 = reinterpret_cast<const v4f*>(charges);
  v4f*       l4 = reinterpret_cast<v4f*>(lds_q);
  const int  nvec = n_atoms >> 2;
  for (int i = tid; i < nvec; i += nthr) {
#if __has_builtin(__builtin_amdgcn_global_load_async_to_lds_b128)
    __builtin_amdgcn_global_load_async_to_lds_b128(
        (as1_v4i*)(q4 + i), (as3_v4i*)(l4 + i), /*offset=*/0, /*cpol=*/0);
#else
    l4[i] = q4[i];
#endif
  }
#if __has_builtin(__builtin_amdgcn_global_load_async_to_lds_b128)
# if __has_builtin(__builtin_amdgcn_s_wait_asynccnt)
  __builtin_amdgcn_s_wait_asynccnt(0);
# else
  asm volatile("s_wait_asynccnt 0" ::: "memory");
# endif
#endif
  __syncthreads();

  // ---- Stream the pair arrays (read-once: non-temporal, 128-bit loads) ----
  const v4f* pd4 = reinterpret_cast<const v4f*>(pair_dist);
  const v4i* pf4 = reinterpret_cast<const v4i*>(pair_first);
  const v4i* ps4 = reinterpret_cast<const v4i*>(pair_second);

  const int nq       = n_pairs >> 2;
  const int stride   = (int)gridDim.x * nthr;
  const float rscale = PI_F / RADIUS_F;

  for (int i = (int)blockIdx.x * nthr + tid; i < nq; i += stride) {
    v4f r4 = __builtin_nontemporal_load(pd4 + i);
    v4i a4 = __builtin_nontemporal_load(pf4 + i);
    v4i b4 = __builtin_nontemporal_load(ps4 + i);

    // prefetch next grid-stride chunk of the streams
    __builtin_prefetch(pd4 + i + stride, 0, 0);
    __builtin_prefetch(pf4 + i + stride, 0, 0);
    __builtin_prefetch(ps4 + i + stride, 0, 0);

    float rr[4] = { r4[0], r4[1], r4[2], r4[3] };
    int   aa[4] = { a4[0], a4[1], a4[2], a4[3] };
    int   bb[4] = { b4[0], b4[1], b4[2], b4[3] };

#pragma unroll
    for (int k = 0; k < 4; ++k) {
      float r  = rr[k];
      int   a  = aa[k];
      int   b  = bb[k];
      float qi = lds_q[a];                      // ds_load_b32 gather
      float qj = lds_q[b];                      // ds_load_b32 gather
      int   m  = mol_index[a];                  // L2-resident gather (256 KB table)
      float scr = 0.5f * (1.0f + __cosf(r * rscale));
      scr = (r < RADIUS_F) ? scr : 0.0f;
      float e = qi * qj * __builtin_amdgcn_rcpf(r) * scr;
      atomicAdd(&lds_bin[m], e);                // ds_add_f32 (no return)
    }
  }

  // scalar tail (n_pairs not multiple of 4; not hit for 2^24 but kept correct)
  for (int i = (nq << 2) + (int)blockIdx.x * nthr + tid; i < n_pairs; i += stride) {
    float r  = pair_dist[i];
    int   a  = pair_first[i];
    int   b  = pair_second[i];
    float qi = lds_q[a];
    float qj = lds_q[b];
    int   m  = mol_index[a];
    float scr = 0.5f * (1.0f + __cosf(r * rscale));
    scr = (r < RADIUS_F) ? scr : 0.0f;
    atomicAdd(&lds_bin[m], qi * qj * __builtin_amdgcn_rcpf(r) * scr);
  }

  __syncthreads();

  // ---- Flush per-workgroup bins to global (one f32 atomic per bin) ----
  const float scale = 0.5f * ECONV_F;
  for (int b = tid; b < n_mol; b += nthr) {
    float v = lds_bin[b];
    atomicAdd(out + b, scale * v);              // global_atomic_add_f32
  }
}

__global__ void zero_out_kernel(float* __restrict__ out, int n) {
  int i = blockIdx.x * blockDim.x + threadIdx.x;
  if (i < n) out[i] = 0.0f;
}

extern "C" void kernel_launch(void* const* d_in, const int* in_sizes, int n_in,
                              void* d_out, int out_size, void* d_ws, size_t ws_size,
                              hipStream_t stream) {
  const float* charges     = (const float*)d_in[0];
  const float* pair_dist   = (const float*)d_in[1];
  const int*   pair_first  = (const int*)d_in[2];
  const int*   pair_second = (const int*)d_in[3];
  const int*   mol_index   = (const int*)d_in[4];
  float* out = (float*)d_out;

  const int n_atoms = in_sizes[0];
  const int n_pairs = in_sizes[1];
  const int n_mol   = out_size;

  // output is accumulated with atomics; harness poisons it, so zero it first
  zero_out_kernel<<<(n_mol + 255) / 256, 256, 0, stream>>>(out, n_mol);

  const size_t shmem = (size_t)(n_atoms + n_mol) * sizeof(float); // 264 KB < 320 KB/WGP
  (void)hipFuncSetAttribute(reinterpret_cast<const void*>(screened_coulomb_kernel),
                            hipFuncAttributeMaxDynamicSharedMemorySize, (int)shmem);

  // 512 workgroups x 1024 threads (32 wave32 waves each); grid-stride over pairs
  screened_coulomb_kernel<<<512, 1024, shmem, stream>>>(
      charges, pair_dist, pair_first, pair_second, mol_index, out,
      n_pairs, n_atoms, n_mol);
}